// SimplifiedMoE_76089640616226
// MI455X (gfx1250) — compile-verified
//
#include <hip/hip_runtime.h>
#include <math.h>

typedef __attribute__((ext_vector_type(2))) float v2f;
typedef __attribute__((ext_vector_type(8))) float v8f;

#define NUM_EXPERTS 8
#define D_IN 128
#define I_SZ 16
#define N_TOK (16 * 4096)
#define WAVES_PER_BLOCK 4
#define BLOCK_THREADS (WAVES_PER_BLOCK * 32)

__device__ __forceinline__ float gelu_exact(float v) {
  // matches jax.nn.gelu(approximate=False): 0.5*v*(1+erf(v/sqrt(2)))
  return 0.5f * v * (1.0f + erff(v * 0.70710678118654752440f));
}

__global__ __launch_bounds__(BLOCK_THREADS)
void moe_wmma_f32_kernel(const float* __restrict__ x,
                         const float* __restrict__ routing,
                         const float* __restrict__ w1,
                         const float* __restrict__ w2,
                         float* __restrict__ agg_out,
                         float* __restrict__ exp_out) {
  // Per-wave scratch: Y_e tiles (16 tok x 16 i) for all experts (cross-lane transpose)
  __shared__ float Ys[WAVES_PER_BLOCK][NUM_EXPERTS][16 * 16];

  const int lane  = threadIdx.x & 31;
  const int wave  = threadIdx.x >> 5;
  const int tile  = blockIdx.x * WAVES_PER_BLOCK + wave;
  const int tok0  = tile * 16;
  const int nrow  = lane & 15;         // A: M row (token) / B: N column
  const int khalf = (lane >> 4) << 1;  // A/B fragment K sub-offset: 0 or 2
  const int hi8   = (lane >> 4) << 3;  // C/D row offset: 0 or 8

  // ---- Preload x tile (16 tokens x 128) as 32 fp32 A-fragments (K chunks of 4)
  v2f a1[32];
  const float* xrow = x + (size_t)(tok0 + nrow) * D_IN + khalf;
#pragma unroll
  for (int kc = 0; kc < 32; ++kc)
    a1[kc] = *(const v2f*)(xrow + kc * 4);

  // ---- Phase 1: Y_e = gelu(x @ w1_e), parked in LDS
#pragma unroll 1
  for (int e = 0; e < NUM_EXPERTS; ++e) {
    const float* w1e = w1 + (size_t)e * D_IN * I_SZ;
    v8f acc = {};
#pragma unroll
    for (int kc = 0; kc < 32; ++kc) {
      const int K = kc * 4 + khalf;
      v2f b;
      b.x = w1e[K * I_SZ + nrow];
      b.y = w1e[(K + 1) * I_SZ + nrow];
      acc = __builtin_amdgcn_wmma_f32_16x16x4_f32(false, a1[kc], false, b,
                                                  (short)0, acc, false, false);
    }
    float* ys = &Ys[wave][e][0];
#pragma unroll
    for (int r = 0; r < 8; ++r)
      ys[(r + hi8) * 16 + nrow] = gelu_exact(acc[r]);
  }

  __syncthreads();

  // ---- Phase 2: Out_e = Y_e @ w2_e -> experts_out stores + routing-weighted agg
  v8f agg[8];
  {
    v8f vzero = {};
#pragma unroll
    for (int m = 0; m < 8; ++m) agg[m] = vzero;
  }

#pragma unroll 1
  for (int e = 0; e < NUM_EXPERTS; ++e) {
    const float* w2e = w2 + (size_t)e * I_SZ * D_IN;
    const float* ys  = &Ys[wave][e][0];

    float rwv[8];
#pragma unroll
    for (int r = 0; r < 8; ++r)
      rwv[r] = routing[(size_t)(tok0 + r + hi8) * NUM_EXPERTS + e];

#pragma unroll
    for (int m = 0; m < 8; ++m) {
      v8f acc2 = {};
#pragma unroll
      for (int kc = 0; kc < 4; ++kc) {
        const int K = kc * 4 + khalf;
        v2f a2 = *(const v2f*)(ys + nrow * 16 + K);  // A frag of Y (from LDS transpose)
        v2f b;
        b.x = w2e[K * D_IN + m * 16 + nrow];
        b.y = w2e[(K + 1) * D_IN + m * 16 + nrow];
        acc2 = __builtin_amdgcn_wmma_f32_16x16x4_f32(false, a2, false, b,
                                                     (short)0, acc2, false, false);
      }
#pragma unroll
      for (int r = 0; r < 8; ++r) {
        const size_t t = (size_t)(tok0 + r + hi8);
        exp_out[t * (NUM_EXPERTS * D_IN) + (size_t)e * D_IN + m * 16 + nrow] = acc2[r];
        agg[m][r] += rwv[r] * acc2[r];
      }
    }
  }

  // ---- Phase 3: store routing-weighted aggregate
#pragma unroll
  for (int m = 0; m < 8; ++m) {
#pragma unroll
    for (int r = 0; r < 8; ++r)
      agg_out[(size_t)(tok0 + r + hi8) * D_IN + m * 16 + nrow] = agg[m][r];
  }
}

extern "C" void kernel_launch(void* const* d_in, const int* in_sizes, int n_in,
                              void* d_out, int out_size, void* d_ws, size_t ws_size,
                              hipStream_t stream) {
  (void)in_sizes; (void)n_in; (void)out_size; (void)d_ws; (void)ws_size;
  const float* x  = (const float*)d_in[0];
  const float* rw = (const float*)d_in[1];
  const float* w1 = (const float*)d_in[2];
  const float* w2 = (const float*)d_in[3];

  float* agg     = (float*)d_out;                       // [B,T,D]  first output
  float* exp_out = agg + (size_t)N_TOK * D_IN;          // [B,T,E,D] second output

  dim3 grid(N_TOK / (16 * WAVES_PER_BLOCK));            // 1024 blocks
  dim3 block(BLOCK_THREADS);                            // 128 threads = 4 waves
  hipLaunchKernelGGL(moe_wmma_f32_kernel, grid, block, 0, stream,
                     x, rw, w1, w2, agg, exp_out);
}